// Skeleton_54314156425466
// MI455X (gfx1250) — compile-verified
//
#include <hip/hip_runtime.h>

// Skeleton forward kinematics: 6D rotations -> world joint positions.
// B=262144, N=16. f32-VALU-bound; data movement via gfx1250 async LDS DMA.

#define TPB 64
#define IN_STRIDE 100   // dwords/element in LDS (96 data + 4 pad; 16B aligned, odd/4 banks)
#define OUT_STRIDE 52   // dwords/element in LDS (48 data + 4 pad)

struct M3 { float a00,a01,a02,a10,a11,a12,a20,a21,a22; };

__device__ __forceinline__ M3 mm(const M3& A, const M3& B) {
  M3 C;
  C.a00 = fmaf(A.a00,B.a00, fmaf(A.a01,B.a10, A.a02*B.a20));
  C.a01 = fmaf(A.a00,B.a01, fmaf(A.a01,B.a11, A.a02*B.a21));
  C.a02 = fmaf(A.a00,B.a02, fmaf(A.a01,B.a12, A.a02*B.a22));
  C.a10 = fmaf(A.a10,B.a00, fmaf(A.a11,B.a10, A.a12*B.a20));
  C.a11 = fmaf(A.a10,B.a01, fmaf(A.a11,B.a11, A.a12*B.a21));
  C.a12 = fmaf(A.a10,B.a02, fmaf(A.a11,B.a12, A.a12*B.a22));
  C.a20 = fmaf(A.a20,B.a00, fmaf(A.a21,B.a10, A.a22*B.a20));
  C.a21 = fmaf(A.a20,B.a01, fmaf(A.a21,B.a11, A.a22*B.a21));
  C.a22 = fmaf(A.a20,B.a02, fmaf(A.a21,B.a12, A.a22*B.a22));
  return C;
}

// Zhou et al. 6D -> 3x3 rotation (rows b1,b2,b3), single v_rsq_f32 per normalize.
__device__ __forceinline__ M3 rot6d(const float* p) {
  float x1=p[0], y1=p[1], z1=p[2], x2=p[3], y2=p[4], z2=p[5];
  float i1 = __builtin_amdgcn_rsqf(fmaf(x1,x1, fmaf(y1,y1, z1*z1)));
  float b1x = x1*i1, b1y = y1*i1, b1z = z1*i1;
  float d = fmaf(b1x,x2, fmaf(b1y,y2, b1z*z2));
  float ux = fmaf(-d,b1x,x2), uy = fmaf(-d,b1y,y2), uz = fmaf(-d,b1z,z2);
  float i2 = __builtin_amdgcn_rsqf(fmaf(ux,ux, fmaf(uy,uy, uz*uz)));
  float b2x = ux*i2, b2y = uy*i2, b2z = uz*i2;
  float b3x = fmaf(b1y,b2z, -b1z*b2y);
  float b3y = fmaf(b1z,b2x, -b1x*b2z);
  float b3z = fmaf(b1x,b2y, -b1y*b2x);
  return {b1x,b1y,b1z, b2x,b2y,b2z, b3x,b3y,b3z};
}

__device__ __forceinline__ void async_load_b128(unsigned lds_off, const void* g) {
  // gfx1250 async DMA: memory -> LDS, tracked by ASYNCcnt, no VGPR writeback.
  asm volatile("global_load_async_to_lds_b128 %0, %1, off"
               :: "v"(lds_off), "v"(g) : "memory");
}
__device__ __forceinline__ void async_store_b128(void* g, unsigned lds_off) {
  asm volatile("global_store_async_from_lds_b128 %0, %1, off"
               :: "v"(g), "v"(lds_off) : "memory");
}
__device__ __forceinline__ void wait_asynccnt0() {
  asm volatile("s_wait_asynccnt 0" ::: "memory");
}

__global__ __launch_bounds__(TPB)
void skel_fk_kernel(const float* __restrict__ angles,
                    const float* __restrict__ xyz,
                    float* __restrict__ out)
{
  __shared__ float s_in [TPB * IN_STRIDE];   // staged angles, padded
  __shared__ float s_out[TPB * OUT_STRIDE];  // staged points, padded

  const int tid = threadIdx.x;
  const float* gin  = angles + (size_t)blockIdx.x * (TPB * 96);
  float*       gout = out    + (size_t)blockIdx.x * (TPB * 48);

  // Low 32 bits of a generic LDS pointer == LDS byte offset (ISA 10.2 aperture rule).
  const unsigned sin_base  = (unsigned)(size_t)(&s_in[0]);
  const unsigned sout_base = (unsigned)(size_t)(&s_out[0]);

  // ---- Stage angles: 24 coalesced B128 async copies per thread (512B/wave-instr) ----
#pragma unroll
  for (int i = 0; i < 24; ++i) {
    unsigned q = (unsigned)(i * TPB + tid) * 4u;  // packed dword index
    unsigned e = q / 96u;                          // element within tile
    unsigned r = q - e * 96u;                      // dword within element
    async_load_b128(sin_base + (e * IN_STRIDE + r) * 4u, (const void*)(gin + q));
  }
  wait_asynccnt0();
  __syncthreads();

  const float* my  = &s_in [tid * IN_STRIDE];
  float*       myo = &s_out[tid * OUT_STRIDE];

  // ---- Kinematic tree, fully unrolled in registers ----
  M3 cur; float cx, cy, cz;

  // root: pw0 = R0 @ xyz[0]   (xyz reads are uniform -> scalar loads)
  cur = rot6d(my);
  {
    float rx = xyz[0], ry = xyz[1], rz = xyz[2];
    cx = fmaf(cur.a00,rx, fmaf(cur.a01,ry, cur.a02*rz));
    cy = fmaf(cur.a10,rx, fmaf(cur.a11,ry, cur.a12*rz));
    cz = fmaf(cur.a20,rx, fmaf(cur.a21,ry, cur.a22*rz));
    myo[0]=cx; myo[1]=cy; myo[2]=cz;
  }

  // edge step: Rw[c] = Rw[p] @ R[c];  pw[c] = Rw[c] @ (xyz[c]-xyz[p]) + pw[p]
  auto step = [&](int c, int p) {
    M3 R = rot6d(my + c * 6);
    cur = mm(cur, R);
    float tx = xyz[3*c+0] - xyz[3*p+0];   // uniform (SGPR) bone vector
    float ty = xyz[3*c+1] - xyz[3*p+1];
    float tz = xyz[3*c+2] - xyz[3*p+2];
    cx = fmaf(cur.a00,tx, fmaf(cur.a01,ty, fmaf(cur.a02,tz, cx)));
    cy = fmaf(cur.a10,tx, fmaf(cur.a11,ty, fmaf(cur.a12,tz, cy)));
    cz = fmaf(cur.a20,tx, fmaf(cur.a21,ty, fmaf(cur.a22,tz, cz)));
    myo[3*c+0]=cx; myo[3*c+1]=cy; myo[3*c+2]=cz;
  };

  step(1,0); step(2,1); step(3,2);
  M3 R3 = cur; float p3x = cx, p3y = cy, p3z = cz;   // branch point (joint 3)
  step(4,3);  step(5,4);
  cur = R3; cx = p3x; cy = p3y; cz = p3z;
  step(6,3);  step(7,6);  step(8,7);  step(9,8);  step(10,9);
  cur = R3; cx = p3x; cy = p3y; cz = p3z;
  step(11,3); step(12,11); step(13,12); step(14,13); step(15,14);

  // ---- Drain points: 12 coalesced B128 async stores per thread ----
  __syncthreads();
#pragma unroll
  for (int i = 0; i < 12; ++i) {
    unsigned q = (unsigned)(i * TPB + tid) * 4u;
    unsigned e = q / 48u;
    unsigned r = q - e * 48u;
    async_store_b128((void*)(gout + q), sout_base + (e * OUT_STRIDE + r) * 4u);
  }
  wait_asynccnt0();   // stores landed before wave exit (S_ENDPGM also waits idle)
}

extern "C" void kernel_launch(void* const* d_in, const int* in_sizes, int n_in,
                              void* d_out, int out_size, void* d_ws, size_t ws_size,
                              hipStream_t stream) {
  const float* angles = (const float*)d_in[0];   // [B,16,6] f32
  const float* xyz    = (const float*)d_in[1];   // [1,16,3] f32
  float* out          = (float*)d_out;           // [B,16,3] f32

  const int B = in_sizes[0] / 96;                // 262144
  const int blocks = B / TPB;                    // B is a multiple of 64
  skel_fk_kernel<<<blocks, TPB, 0, stream>>>(angles, xyz, out);
}